// GNN_79061757984919
// MI455X (gfx1250) — compile-verified
//
#include <hip/hip_runtime.h>
#include <math.h>

// MI455X / gfx1250, wave32. Matrix ops via V_WMMA_F32_16X16X4_F32 (fp32 in/out,
// matches reference precision; compute is negligible vs the L2-resident gather
// traffic: node_emb (102 MB) fits in the 192 MB L2, ~0.6 GB of gathers dominate).
// Round-2 change: cap gemm_k256 unroll at 8 to keep the wave under 256 VGPRs
// (round-1 asm showed s_set_vgpr_msb / v256+ operands => occupancy collapse).

typedef __attribute__((ext_vector_type(2))) float v2f;
typedef __attribute__((ext_vector_type(8))) float v8f;

#define N_NB    16
#define EMB     128
#define N_TYPES 4
#define N_RELC  16
#define BATCH   4096
// A-tile row stride in floats: 256 K-columns + 4 pad.
// 260*4B = 1040B keeps 16B alignment and makes row r start at bank (4r mod 64),
// so the 16 rows read by a WMMA A-fragment hit 16 distinct banks.
#define AROW    260

// ---------------------------------------------------------------------------
// Kernel 1: rsim[i][j] = |cos(rela_emb[i], rela_emb[j])|, 16x16. One block.
// ---------------------------------------------------------------------------
__global__ void rsim_kernel(const float* __restrict__ rela_emb,
                            float* __restrict__ rsim) {
  const int t = threadIdx.x;            // 256 threads: t = i*16 + j
  const int i = t >> 4, j = t & 15;
  float dot = 0.f, ni = 0.f, nj = 0.f;
  for (int k = 0; k < EMB; ++k) {
    const float a = rela_emb[i * EMB + k];
    const float b = rela_emb[j * EMB + k];
    dot = fmaf(a, b, dot);
    ni  = fmaf(a, a, ni);
    nj  = fmaf(b, b, nj);
  }
  const float d = fmaxf(sqrtf(ni), 1e-8f) * fmaxf(sqrtf(nj), 1e-8f);
  rsim[t] = fabsf(dot / d);
}

// ---------------------------------------------------------------------------
// Fused K=256 WMMA GEMM for one 16x16 output tile:
//   C = A[:,0:128] @ Wself + A[:,128:256] @ Wnb
// Arow = LDS pointer to this lane's A row (row = lane&15).
// A fragment layout (ISA 16x4 f32): lanes 0-15 hold K = {4k,4k+1},
// lanes 16-31 hold K = {4k+2,4k+3} of the same rows; B mirrors it.
// Unroll capped at 8: ~16 in-flight B loads per wave (enough to cover L2
// latency) without blowing past 256 VGPRs.
// ---------------------------------------------------------------------------
__device__ __forceinline__ v8f gemm_k256(const float* __restrict__ Arow,
                                         const float* __restrict__ Wself,
                                         const float* __restrict__ Wnb,
                                         int n, int half) {
  v8f c = {};
#pragma unroll 8
  for (int kk = 0; kk < 32; ++kk) {            // K = 0..127  -> Wself
    const int k0 = kk * 4 + half * 2;
    v2f a = *(const v2f*)(Arow + k0);          // 8B-aligned ds_load_b64
    v2f w;
    w.x = Wself[k0 * EMB + n];
    w.y = Wself[(k0 + 1) * EMB + n];
    c = __builtin_amdgcn_wmma_f32_16x16x4_f32(false, a, false, w,
                                              (short)0, c, false, false);
  }
#pragma unroll 8
  for (int kk = 0; kk < 32; ++kk) {            // K = 128..255 -> Wnb
    const int k0 = kk * 4 + half * 2;
    v2f a = *(const v2f*)(Arow + EMB + k0);
    v2f w;
    w.x = Wnb[k0 * EMB + n];
    w.y = Wnb[(k0 + 1) * EMB + n];
    c = __builtin_amdgcn_wmma_f32_16x16x4_f32(false, a, false, w,
                                              (short)0, c, false, false);
  }
  return c;
}

// ---------------------------------------------------------------------------
// Kernel 2 (hop 1): one wave per batch element b. Its 16 first-hop neighbors
// form one 16-row M-tile. v1[b, i, :] = relu(ev1 @ Ws0 + agg(ev2) @ Wn0).
// Block = 128 threads = 4 waves = 4 batch elements.
// ---------------------------------------------------------------------------
__global__ __launch_bounds__(128) void hop1_kernel(
    const int* __restrict__ node, const int* __restrict__ relation,
    const int* __restrict__ adj_node, const int* __restrict__ adj_rela,
    const int* __restrict__ node_type, const float* __restrict__ node_emb,
    const float* __restrict__ rela_emb, const float* __restrict__ type_w,
    const float* __restrict__ Ws0, const float* __restrict__ Wn0,
    const float* __restrict__ rsim, float* __restrict__ v1) {
  extern __shared__ float smem[];
  const int lane = threadIdx.x & 31;
  const int wave = threadIdx.x >> 5;
  const int b    = blockIdx.x * 4 + wave;
  float* A = smem + wave * (16 * AROW);        // this wave's 16x256 A-tile

  const int rel = relation[b];
  const int nd0 = node[b];
  const int col = lane * 4;                    // lane owns cols [col, col+4)
  const float4 rv = *(const float4*)(rela_emb + rel * EMB + col);
  const float* __restrict__ rs = rsim + rel * N_RELC;

  // Build A = [ self=node_emb[e1]*rel_vec | agg=softmax-wsum(node_emb[e2])*rel_vec ]
  for (int i = 0; i < N_NB; ++i) {
    const int e1i = adj_node[nd0 * N_NB + i];
    const float* tw = type_w + node_type[e1i] * N_TYPES;
    float sc[N_NB];
    int e2[N_NB];
    float m = -3.0e38f;
#pragma unroll
    for (int k = 0; k < N_NB; ++k) {           // uniform (broadcast) loads
      const int e2k = adj_node[e1i * N_NB + k];
      const int r2k = adj_rela[e1i * N_NB + k];
      e2[k] = e2k;
      const float s = rs[r2k] * tw[node_type[e2k]];
      sc[k] = s;
      m = fmaxf(m, s);
    }
    float den = 0.f;
#pragma unroll
    for (int k = 0; k < N_NB; ++k) { sc[k] = __expf(sc[k] - m); den += sc[k]; }
    const float inv = 1.f / den;

    float4 acc = make_float4(0.f, 0.f, 0.f, 0.f);
#pragma unroll
    for (int k = 0; k < N_NB; ++k) {           // coalesced 512B row gathers
      const float w = sc[k] * inv;
      const float4 nv = *(const float4*)(node_emb + (size_t)e2[k] * EMB + col);
      acc.x = fmaf(w, nv.x, acc.x);
      acc.y = fmaf(w, nv.y, acc.y);
      acc.z = fmaf(w, nv.z, acc.z);
      acc.w = fmaf(w, nv.w, acc.w);
    }
    const float4 sv = *(const float4*)(node_emb + (size_t)e1i * EMB + col);
    float* Ar = A + i * AROW;
    *(float4*)(Ar + col) =
        make_float4(sv.x * rv.x, sv.y * rv.y, sv.z * rv.z, sv.w * rv.w);
    *(float4*)(Ar + EMB + col) =
        make_float4(acc.x * rv.x, acc.y * rv.y, acc.z * rv.z, acc.w * rv.w);
  }
  __syncthreads();

  // WMMA GEMM: [16 x 256] @ [Ws0 ; Wn0] -> relu -> v1[b]
  const int half = lane >> 4;
  const int l16  = lane & 15;
  float* __restrict__ dst = v1 + (size_t)b * N_NB * EMB;
  for (int nt = 0; nt < 8; ++nt) {
    const int n = nt * 16 + l16;
    v8f c = gemm_k256(A + l16 * AROW, Ws0, Wn0, n, half);
#pragma unroll
    for (int r = 0; r < 8; ++r)
      dst[(r + half * 8) * EMB + n] = fmaxf(c[r], 0.f);
  }
}

// ---------------------------------------------------------------------------
// Kernel 3 (hop 0 + final): one wave per 16 consecutive batch rows.
//   v0  = relu([ev0 | agg0(ev1)] @ [Ws0;Wn0])
//   out = [v0 | agg00(v1)] @ [Ws1;Wn1], then row-wise L2 normalize.
// Block = 128 threads = 4 waves = 64 batch rows.
// ---------------------------------------------------------------------------
__global__ __launch_bounds__(128) void hop0_kernel(
    const int* __restrict__ node, const int* __restrict__ relation,
    const int* __restrict__ adj_node, const int* __restrict__ adj_rela,
    const int* __restrict__ node_type, const float* __restrict__ node_emb,
    const float* __restrict__ rela_emb, const float* __restrict__ type_w,
    const float* __restrict__ Ws0, const float* __restrict__ Wn0,
    const float* __restrict__ Ws1, const float* __restrict__ Wn1,
    const float* __restrict__ rsim, const float* __restrict__ v1,
    float* __restrict__ out) {
  extern __shared__ float smem[];
  const int lane = threadIdx.x & 31;
  const int wave = threadIdx.x >> 5;
  const int WSZ  = 16 * AROW + 16 * 16;        // A-tile + saved softmax weights
  float* A  = smem + wave * WSZ;
  float* sW = A + 16 * AROW;                   // w00[16][16]
  const int tile = blockIdx.x * 4 + wave;      // rows r=0..15 <-> b = tile*16+r
  const int col  = lane * 4;
  const int half = lane >> 4;
  const int l16  = lane & 15;

  // ---- Phase 1: A = [ev0 | agg0], record softmax weights -------------------
  for (int r = 0; r < 16; ++r) {
    const int bb  = tile * 16 + r;
    const int rel = relation[bb];
    const int nd  = node[bb];
    const float* tw = type_w + node_type[nd] * N_TYPES;
    const float* rs = rsim + rel * N_RELC;
    float sc[N_NB];
    int e1[N_NB];
    float m = -3.0e38f;
#pragma unroll
    for (int k = 0; k < N_NB; ++k) {
      const int e1k = adj_node[nd * N_NB + k];
      const int r1k = adj_rela[nd * N_NB + k];
      e1[k] = e1k;
      const float s = rs[r1k] * tw[node_type[e1k]];
      sc[k] = s;
      m = fmaxf(m, s);
    }
    float den = 0.f;
#pragma unroll
    for (int k = 0; k < N_NB; ++k) { sc[k] = __expf(sc[k] - m); den += sc[k]; }
    const float inv = 1.f / den;
#pragma unroll
    for (int k = 0; k < N_NB; ++k)             // uniform value, same-addr store
      sW[r * 16 + k] = sc[k] * inv;

    const float4 rv = *(const float4*)(rela_emb + rel * EMB + col);
    float4 acc = make_float4(0.f, 0.f, 0.f, 0.f);
#pragma unroll
    for (int k = 0; k < N_NB; ++k) {
      const float w = sc[k] * inv;
      const float4 nv = *(const float4*)(node_emb + (size_t)e1[k] * EMB + col);
      acc.x = fmaf(w, nv.x, acc.x);
      acc.y = fmaf(w, nv.y, acc.y);
      acc.z = fmaf(w, nv.z, acc.z);
      acc.w = fmaf(w, nv.w, acc.w);
    }
    const float4 sv = *(const float4*)(node_emb + (size_t)nd * EMB + col);
    float* Ar = A + r * AROW;
    *(float4*)(Ar + col) =
        make_float4(sv.x * rv.x, sv.y * rv.y, sv.z * rv.z, sv.w * rv.w);
    *(float4*)(Ar + EMB + col) =
        make_float4(acc.x * rv.x, acc.y * rv.y, acc.z * rv.z, acc.w * rv.w);
  }
  __syncthreads();

  // ---- Phase 2: GEMM-1 (Ws0/Wn0), relu, rebuild A = [v0 | agg(v1)] --------
  v8f cb[8];
  for (int nt = 0; nt < 8; ++nt)
    cb[nt] = gemm_k256(A + l16 * AROW, Ws0, Wn0, nt * 16 + l16, half);
  __syncthreads();
#pragma unroll
  for (int nt = 0; nt < 8; ++nt)
#pragma unroll
    for (int r = 0; r < 8; ++r)
      A[(r + half * 8) * AROW + nt * 16 + l16] = fmaxf(cb[nt][r], 0.f);

  for (int r = 0; r < 16; ++r) {
    const int bb = tile * 16 + r;
    float4 acc = make_float4(0.f, 0.f, 0.f, 0.f);
#pragma unroll
    for (int k = 0; k < N_NB; ++k) {
      const float w = sW[r * 16 + k];
      const float4 nv = *(const float4*)(v1 + ((size_t)bb * N_NB + k) * EMB + col);
      acc.x = fmaf(w, nv.x, acc.x);
      acc.y = fmaf(w, nv.y, acc.y);
      acc.z = fmaf(w, nv.z, acc.z);
      acc.w = fmaf(w, nv.w, acc.w);
    }
    *(float4*)(A + r * AROW + EMB + col) = acc;
  }
  __syncthreads();

  // ---- Phase 3: GEMM-2 (Ws1/Wn1) ------------------------------------------
  for (int nt = 0; nt < 8; ++nt)
    cb[nt] = gemm_k256(A + l16 * AROW, Ws1, Wn1, nt * 16 + l16, half);

  // ---- Phase 4: row-wise L2 normalize (16-lane butterfly) + store ---------
#pragma unroll
  for (int r = 0; r < 8; ++r) {
    float s = 0.f;
#pragma unroll
    for (int nt = 0; nt < 8; ++nt) s = fmaf(cb[nt][r], cb[nt][r], s);
    s += __shfl_xor(s, 1, 32);                 // masks < 16: stays in half
    s += __shfl_xor(s, 2, 32);
    s += __shfl_xor(s, 4, 32);
    s += __shfl_xor(s, 8, 32);
    const float invn = 1.f / fmaxf(sqrtf(s), 1e-12f);
    const int row = r + half * 8;
    const int bb  = tile * 16 + row;
#pragma unroll
    for (int nt = 0; nt < 8; ++nt)
      out[(size_t)bb * EMB + nt * 16 + l16] = cb[nt][r] * invn;
  }
}

// ---------------------------------------------------------------------------
extern "C" void kernel_launch(void* const* d_in, const int* in_sizes, int n_in,
                              void* d_out, int out_size, void* d_ws,
                              size_t ws_size, hipStream_t stream) {
  (void)in_sizes; (void)n_in; (void)out_size; (void)ws_size;
  const int*   node      = (const int*)  d_in[0];
  const int*   relation  = (const int*)  d_in[1];
  const int*   adj_node  = (const int*)  d_in[2];
  const int*   adj_rela  = (const int*)  d_in[3];
  const int*   node_type = (const int*)  d_in[4];
  const float* node_emb  = (const float*)d_in[5];
  const float* rela_emb  = (const float*)d_in[6];
  const float* type_w    = (const float*)d_in[7];
  const float* Ws0       = (const float*)d_in[8];
  const float* Wn0       = (const float*)d_in[9];
  const float* Ws1       = (const float*)d_in[10];
  const float* Wn1       = (const float*)d_in[11];
  float* out  = (float*)d_out;

  float* ws   = (float*)d_ws;
  float* rsim = ws;                      // 256 floats
  float* v1   = ws + 1024;               // B*16*128 floats = 32 MiB

  rsim_kernel<<<1, 256, 0, stream>>>(rela_emb, rsim);

  const size_t sh2 = (size_t)4 * 16 * AROW * sizeof(float);       // ~65 KB
  hop1_kernel<<<BATCH / 4, 128, sh2, stream>>>(
      node, relation, adj_node, adj_rela, node_type, node_emb, rela_emb,
      type_w, Ws0, Wn0, rsim, v1);

  const size_t sh3 = (size_t)4 * (16 * AROW + 256) * sizeof(float); // ~69 KB
  hop0_kernel<<<BATCH / 64, 128, sh3, stream>>>(
      node, relation, adj_node, adj_rela, node_type, node_emb, rela_emb,
      type_w, Ws0, Wn0, Ws1, Wn1, rsim, v1, out);
}